// InnerFaceShiftTriple_41970420416942
// MI455X (gfx1250) — compile-verified
//
#include <hip/hip_runtime.h>
#include <float.h>

typedef float v2f __attribute__((ext_vector_type(2)));
typedef float v8f __attribute__((ext_vector_type(8)));

#define HW 9216   // 96*96
#define NM 2304   // 48*48 hole pixels
#define CF 64     // C/2
#define NB 4      // batch

// ---------------- Kernel 1: init output: copy x into ch 0..127, zero ch 128..191
__global__ void init_out_kernel(const float* __restrict__ x, float* __restrict__ out) {
  size_t i = (size_t)blockIdx.x * blockDim.x + threadIdx.x;
  const size_t total = (size_t)NB * 192 * HW;
  if (i >= total) return;
  int pix = (int)(i % HW);
  int ch  = (int)((i / HW) % 192);
  int b   = (int)(i / ((size_t)192 * HW));
  out[i] = (ch < 128) ? x[((size_t)b * 128 + ch) * HW + pix] : 0.0f;
}

// ---------------- Kernel 2: per-pixel inverse norms + fused masked column {scale,bias}
// sbo[j] = flag==0  ? {1/||latter_j||, 0} : {0, -FLT_MAX}
// sbf[j] = flagf==0 ? {1/||flip_j||,   0} : {0, -FLT_MAX}
// so the masked scaled cosine is a single fma: v = (dot*rnf_row)*w.x + w.y
__global__ void norms_kernel(const float* __restrict__ x, const float* __restrict__ flip,
                             const int* __restrict__ flag, const int* __restrict__ flagf,
                             float* __restrict__ rnf, float2* __restrict__ sbo,
                             float2* __restrict__ sbf) {
  int i = blockIdx.x * blockDim.x + threadIdx.x;
  if (i >= NB * HW) return;
  int b = i / HW, pix = i % HW;
  const float* xf = x + (size_t)b * 128 * HW + pix;
  const float* xl = xf + (size_t)CF * HW;
  const float* ff = flip + (size_t)b * CF * HW + pix;
  float sf = 0.f, sl = 0.f, sfl = 0.f;
  #pragma unroll 4
  for (int c = 0; c < CF; ++c) {
    float a = xf[(size_t)c * HW]; sf  += a * a;
    float d = xl[(size_t)c * HW]; sl  += d * d;
    float e = ff[(size_t)c * HW]; sfl += e * e;
  }
  rnf[i] = rsqrtf(sf);
  float rl = rsqrtf(sl), rfl = rsqrtf(sfl);
  bool oko = (flag[i] == 0);
  bool okf = (flagf[i] == 0);
  sbo[i] = oko ? make_float2(rl, 0.f)  : make_float2(0.f, -FLT_MAX);
  sbf[i] = okf ? make_float2(rfl, 0.f) : make_float2(0.f, -FLT_MAX);
}

// ---------------- Kernel 3: fused cosine GEMM (fp32 WMMA) + mask + row max/argmax
// grid = (NM/16 row tiles, 2 branches, NB batches), block = 128 (4 waves)
__launch_bounds__(128)
__global__ void cos_argmax_kernel(const float* __restrict__ x,
                                  const float* __restrict__ flip,
                                  const int* __restrict__ pidx,
                                  const int* __restrict__ qidx,
                                  const float* __restrict__ rnf,
                                  const float2* __restrict__ sbo,
                                  const float2* __restrict__ sbf,
                                  float* __restrict__ maxout,
                                  int* __restrict__ bestout) {
  __shared__ int   rowidx_s[16];
  __shared__ float rnfs[16];
  __shared__ float Alds[16][66];                     // 16 rows x 64 K, padded (bank spread)
  __shared__ __align__(16) float Blds[4][64][16];    // per-wave B slab: 64 K x 16 N
  __shared__ float redv[4][16];
  __shared__ int   redi[4][16];

  const int tile = blockIdx.x;
  const int br   = blockIdx.y;
  const int b    = blockIdx.z;
  const int tid  = threadIdx.x;
  const int wave = tid >> 5;
  const int lane = tid & 31;
  const int half = lane >> 4;   // wave32: lanes 0-15 vs 16-31
  const int l16  = lane & 15;

  const int*    rows = ((br == 0) ? pidx : qidx) + b * NM;
  const float*  Bsrc = (br == 0) ? (x + ((size_t)b * 128 + CF) * HW)
                                 : (flip + (size_t)b * CF * HW);
  const float2* csb  = ((br == 0) ? sbo : sbf) + b * HW;

  if (tid < 16) {
    int r = rows[tile * 16 + tid];
    rowidx_s[tid] = r;
    rnfs[tid] = rnf[b * HW + r];
  }
  __syncthreads();

  // gather A tile: A[m][k] = former[k][hole_row m]
  for (int i = tid; i < 16 * 64; i += 128) {
    int r = i >> 6, k = i & 63;
    Alds[r][k] = x[((size_t)b * 128 + k) * HW + rowidx_s[r]];
  }
  __syncthreads();

  // A fragments for all 16 K-steps (f32 16x4 layout: lanes0-15 K{0,1}, lanes16-31 K{2,3})
  v2f afrag[16];
  #pragma unroll
  for (int s = 0; s < 16; ++s) {
    int kb = 4 * s + 2 * half;
    afrag[s].x = Alds[l16][kb];
    afrag[s].y = Alds[l16][kb + 1];
  }
  float rnfreg[8];
  #pragma unroll
  for (int e = 0; e < 8; ++e) rnfreg[e] = rnfs[e + 8 * half];  // C row = e + 8*half

  float bmax[8]; int bidx[8];
  #pragma unroll
  for (int e = 0; e < 8; ++e) { bmax[e] = -FLT_MAX; bidx[e] = 0; }

  for (int ct = wave; ct < HW / 16; ct += 4) {
    const int colbase = ct * 16;
    // stage B tile (64x16 f32, contiguous 64B rows) into this wave's private LDS slab
    #pragma unroll
    for (int r = 0; r < 2; ++r) {
      const int k = lane * 2 + r;
      const float4* src = reinterpret_cast<const float4*>(Bsrc + (size_t)k * HW + colbase);
      float4* dst = reinterpret_cast<float4*>(&Blds[wave][k][0]);
      dst[0] = src[0]; dst[1] = src[1]; dst[2] = src[2]; dst[3] = src[3];
    }
    if (ct + 4 < HW / 16)
      __builtin_prefetch(Bsrc + (size_t)(lane * 2) * HW + (size_t)(ct + 4) * 16, 0, 1);

    const int col = colbase + l16;
    const float2 w = csb[col];   // {masked col scale, bias (-FLT_MAX when masked)}

    v8f c = {};
    #pragma unroll
    for (int s = 0; s < 16; ++s) {
      const int kb = 4 * s + 2 * half;
      v2f bf;
      bf.x = Blds[wave][kb][l16];
      bf.y = Blds[wave][kb + 1][l16];
      // D = A(16x4 f32) x B(4x16 f32) + C, full fp32
      c = __builtin_amdgcn_wmma_f32_16x16x4_f32(false, afrag[s], false, bf,
                                                (short)0, c, false, false);
    }

    // masked scaled cosine + running max/argmax.
    // Each lane's columns arrive in increasing order, so strict '>' keeps the
    // first (lowest-index) maximum -- no in-loop index tie-break needed.
    #pragma unroll
    for (int e = 0; e < 8; ++e) {
      float v = fmaf(c[e] * rnfreg[e], w.x, w.y);
      if (v > bmax[e]) { bmax[e] = v; bidx[e] = col; }
    }
  }

  // reduce over the 16 lanes holding the same C row (xor stays within 16-lane groups)
  #pragma unroll
  for (int e = 0; e < 8; ++e) {
    float v = bmax[e]; int j = bidx[e];
    #pragma unroll
    for (int off = 1; off < 16; off <<= 1) {
      float ov = __shfl_xor(v, off, 32);
      int   oj = __shfl_xor(j, off, 32);
      if (ov > v || (ov == v && oj < j)) { v = ov; j = oj; }
    }
    if (l16 == 0) { redv[wave][e + 8 * half] = v; redi[wave][e + 8 * half] = j; }
  }
  __syncthreads();

  if (tid < 16) {
    float v = redv[0][tid]; int j = redi[0][tid];
    #pragma unroll
    for (int w2 = 1; w2 < 4; ++w2) {
      float ov = redv[w2][tid]; int oj = redi[w2][tid];
      if (ov > v || (ov == v && oj < j)) { v = ov; j = oj; }
    }
    const int gm = tile * 16 + tid;
    maxout[((size_t)b * 2 + br) * NM + gm] = v;
    bestout[((size_t)b * 2 + br) * NM + gm] = j;
  }
}

// ---------------- Kernel 4: branch decision + dedup'ed argmax column marks
__global__ void decide_kernel(const float* __restrict__ maxv, const int* __restrict__ best,
                              int* __restrict__ selo, int* __restrict__ colmark) {
  int i = blockIdx.x * blockDim.x + threadIdx.x;
  if (i >= NB * NM) return;
  int b = i / NM, m = i % NM;
  float mo = maxv[((size_t)b * 2 + 0) * NM + m];
  float mf = maxv[((size_t)b * 2 + 1) * NM + m];
  selo[i] = (mo >= mf) ? 1 : 0;
  // col sets are built from ALL rows' argmaxes (reference quirk), dedup via idempotent store
  colmark[((size_t)b * 2 + 0) * HW + best[((size_t)b * 2 + 0) * NM + m]] = 1;
  colmark[((size_t)b * 2 + 1) * HW + best[((size_t)b * 2 + 1) * NM + m]] = 1;
}

// ---------------- Kernel 5: S_o[b][c] / S_f[b][c] masked column sums
__global__ void sums_kernel(const float* __restrict__ x, const float* __restrict__ flip,
                            const int* __restrict__ colmark, float* __restrict__ S) {
  __shared__ float sh[2][256];
  const int c = blockIdx.x;   // 0..63
  const int b = blockIdx.y;   // 0..3
  const int tid = threadIdx.x;
  const float* lat = x + ((size_t)b * 128 + CF + c) * HW;
  const float* fp  = flip + ((size_t)b * CF + c) * HW;
  const int* mo = colmark + ((size_t)b * 2 + 0) * HW;
  const int* mf = colmark + ((size_t)b * 2 + 1) * HW;
  float ao = 0.f, af = 0.f;
  for (int j = tid; j < HW; j += 256) {
    if (mo[j]) ao += lat[j];
    if (mf[j]) af += fp[j];
  }
  sh[0][tid] = ao; sh[1][tid] = af;
  __syncthreads();
  for (int s = 128; s > 0; s >>= 1) {
    if (tid < s) { sh[0][tid] += sh[0][tid + s]; sh[1][tid] += sh[1][tid + s]; }
    __syncthreads();
  }
  if (tid == 0) {
    S[((size_t)b * 2 + 0) * CF + c] = sh[0][0];
    S[((size_t)b * 2 + 1) * CF + c] = sh[1][0];
  }
}

// ---------------- Kernel 6: scatter S_o/S_f into hole positions of shifted channels
__global__ void scatter_kernel(const int* __restrict__ pidx, const int* __restrict__ selo,
                               const float* __restrict__ S, float* __restrict__ out) {
  const int bid = blockIdx.x;          // 0..NB*NM-1
  const int b = bid / NM, m = bid % NM;
  const int c = threadIdx.x;           // 0..63
  const int pos = pidx[b * NM + m];
  const float v = selo[bid] ? S[((size_t)b * 2 + 0) * CF + c]
                            : S[((size_t)b * 2 + 1) * CF + c];
  out[((size_t)b * 192 + 128 + c) * HW + pos] = v;
}

extern "C" void kernel_launch(void* const* d_in, const int* in_sizes, int n_in,
                              void* d_out, int out_size, void* d_ws, size_t ws_size,
                              hipStream_t stream) {
  const float* x     = (const float*)d_in[0];
  const float* flip  = (const float*)d_in[1];
  const int*   flag  = (const int*)d_in[2];
  const int*   flagf = (const int*)d_in[3];
  const int*   pidx  = (const int*)d_in[4];
  const int*   qidx  = (const int*)d_in[5];
  float* out = (float*)d_out;

  // float2 arrays first to keep 8-byte alignment of the workspace carve-out
  char* w = (char*)d_ws;
  float2* sbo    = (float2*)w; w += (size_t)NB * HW * sizeof(float2);
  float2* sbf    = (float2*)w; w += (size_t)NB * HW * sizeof(float2);
  float* rnf     = (float*)w;  w += (size_t)NB * HW * sizeof(float);
  float* maxv    = (float*)w;  w += (size_t)NB * 2 * NM * sizeof(float);
  int*   best    = (int*)w;    w += (size_t)NB * 2 * NM * sizeof(int);
  int*   colmark = (int*)w;    w += (size_t)NB * 2 * HW * sizeof(int);
  int*   selo    = (int*)w;    w += (size_t)NB * NM * sizeof(int);
  float* S       = (float*)w;  w += (size_t)NB * 2 * CF * sizeof(float);

  const size_t total_out = (size_t)NB * 192 * HW;
  init_out_kernel<<<(unsigned)((total_out + 255) / 256), 256, 0, stream>>>(x, out);
  norms_kernel<<<(NB * HW + 255) / 256, 256, 0, stream>>>(x, flip, flag, flagf,
                                                          rnf, sbo, sbf);
  hipMemsetAsync(colmark, 0, (size_t)NB * 2 * HW * sizeof(int), stream);

  dim3 g3(NM / 16, 2, NB);
  cos_argmax_kernel<<<g3, 128, 0, stream>>>(x, flip, pidx, qidx,
                                            rnf, sbo, sbf, maxv, best);
  decide_kernel<<<(NB * NM + 255) / 256, 256, 0, stream>>>(maxv, best, selo, colmark);
  dim3 g5(CF, NB);
  sums_kernel<<<g5, 256, 0, stream>>>(x, flip, colmark, S);
  scatter_kernel<<<NB * NM, CF, 0, stream>>>(pidx, selo, S, out);
}